// LFQ_4415226380689
// MI455X (gfx1250) — compile-verified
//
#include <hip/hip_runtime.h>
#include <hip/hip_bf16.h>
#include <math.h>

// ---------------------------------------------------------------------------
// LFQ forward on gfx1250.
//   d_in: 0=x[8,1024,512] f32, 1=w_in[14,512], 2=b_in[14], 3=w_out[512,14], 4=b_out[512]
//   d_out: out[8*1024*512] f32 | indices[8192] (as float) | aux_loss[1]
//   ws layout (floats): avg[16384] | scalars[2] (+14 pad) | PL[8192*128] | PR[8192*128]
//
// Math restructuring: softmax over the {-1,+1}^14 codebook factorizes into 14
// Bernoullis => per-sample entropy = sum of binary entropies; splitting dims
// 7/7, each token's 16384 probs = pL(128) (outer) pR(128), so
// avg_prob = (1/N) PL^T PR  -> a 128 x 8192 x 128 fp32 GEMM done with
// V_WMMA_F32_16X16X4_F32 (keeps reference fp32 precision).
// ---------------------------------------------------------------------------

#define LFQ_DIM   512
#define LFQ_CDIM  14
#define LFQ_NTOK  8192
#define LFQ_K     16384

typedef float v2f __attribute__((ext_vector_type(2)));
typedef float v8f __attribute__((ext_vector_type(8)));
typedef unsigned int u32x4 __attribute__((ext_vector_type(4)));
typedef int i32x4 __attribute__((ext_vector_type(4)));
typedef int i32x8 __attribute__((ext_vector_type(8)));

// Address-space-qualified pointer types for the async-to-LDS builtin
// (param types per compiler diagnostic: v4i in AS1 / AS3).
typedef __attribute__((address_space(1))) i32x4 as1_i32x4;
typedef __attribute__((address_space(3))) i32x4 as3_i32x4;

// CDNA5-specific data movers, guarded so a missing builtin falls back cleanly.
#if defined(__HIP_DEVICE_COMPILE__) && __has_builtin(__builtin_amdgcn_tensor_load_to_lds) && __has_builtin(__builtin_amdgcn_s_wait_tensorcnt)
#define LFQ_TDM 1
#else
#define LFQ_TDM 0
#endif

#if defined(__HIP_DEVICE_COMPILE__) && __has_builtin(__builtin_amdgcn_global_load_async_to_lds_b128) && __has_builtin(__builtin_amdgcn_s_wait_asynccnt)
#define LFQ_ASYNC 1
#else
#define LFQ_ASYNC 0
#endif

#if LFQ_TDM
// 1-D TDM copy of n fp32 elements global -> LDS (n < 65536 so it fits tile_dim0).
// D# per CDNA5 ISA: group0 = {count|flags, lds_addr, ga_lo,
// ga_hi|type=2}; group1 = {data_size=4B, tensor_dim0, tensor_dim1=1,
// tile_dim0=n, tile_dim1=1, stride0=n, ...}. Toolchain uses the 6-arg builtin
// (groups 2/3 + trailing group zero-filled, then cpol).
__device__ __forceinline__ void lfq_tdm_load_1d(float* ldst, const float* gsrc, int n) {
    unsigned lds_off = (unsigned)(unsigned long long)
        (__attribute__((address_space(3))) float*)ldst;
    unsigned long long ga = (unsigned long long)gsrc;
    u32x4 g0;
    g0.x = 1u;                                                  // count=1 (user D#)
    g0.y = lds_off;                                             // lds_addr [63:32]
    g0.z = (unsigned)(ga & 0xFFFFFFFFull);                      // global_addr[31:0]
    g0.w = (unsigned)((ga >> 32) & 0x01FFFFFFull) | (2u << 30); // [56:32] | type=2
    i32x8 g1;
    g1[0] = 2 << 16;                       // workgroup_mask=0, data_size=2 (4 bytes)
    g1[1] = (n & 0xFFFF) << 16;            // tensor_dim0[15:0] -> bits[63:48]
    g1[2] = ((n >> 16) & 0xFFFF) | (1 << 16); // tensor_dim0 hi | tensor_dim1=1 lo
    g1[3] = (n & 0xFFFF) << 16;            // tensor_dim1 hi=0 | tile_dim0=n
    g1[4] = 1;                             // tile_dim1=1, tile_dim2=0
    g1[5] = n;                             // tensor_dim0_stride lo32
    g1[6] = 0;                             // stride0 hi | stride1 lo
    g1[7] = 0;
    i32x4 z4 = {0, 0, 0, 0};
    i32x8 z8 = {0, 0, 0, 0, 0, 0, 0, 0};
    __builtin_amdgcn_tensor_load_to_lds(g0, g1, z4, z4, z8, 0);
}
#endif

// ------------------------- kernel 0: zero ws accumulators -------------------
__global__ void lfq_zero_kernel(float* ws) {
    int i = blockIdx.x * blockDim.x + threadIdx.x;
    if (i < LFQ_K + 2) ws[i] = 0.0f;
}

// ------------------------- kernel 1: fused per-token pass -------------------
// Two threads per token (halves the serial FMA chains, 512 waves in flight):
//   half h computes K-range [256h,256h+256) of project_in, dims [7h,7h+7) of
//   the Bernoulli factorization (half0 -> pL row, half1 -> pR row), and
//   d-range [256h,256h+256) of project_out. 64 blocks x 256 thr = 2*8192.
__global__ __launch_bounds__(256) void lfq_main_kernel(
    const float* __restrict__ x, const float* __restrict__ w_in,
    const float* __restrict__ b_in, const float* __restrict__ w_out,
    const float* __restrict__ b_out,
    float* __restrict__ out, float* __restrict__ idx_out,
    float* __restrict__ PL, float* __restrict__ PR,
    float* __restrict__ scalars) {
    __shared__ float sWin[LFQ_CDIM * LFQ_DIM];   // [c][i]
    __shared__ float sWout[LFQ_DIM * LFQ_CDIM];  // [d][c]
    __shared__ float sBout[LFQ_DIM];
    __shared__ float sBin[16];
    __shared__ float sY[128 * 16];               // per-token exchange (pad 14->16)
    __shared__ float sEnt, sCom;

    const int tid = threadIdx.x;

    // ---- stage weights into LDS (TDM if available) ------------------------
#if LFQ_TDM
    if (tid == 0) {
        lfq_tdm_load_1d(sWin, w_in, LFQ_CDIM * LFQ_DIM);
        lfq_tdm_load_1d(sWout, w_out, LFQ_DIM * LFQ_CDIM);
        __builtin_amdgcn_s_wait_tensorcnt(0);
    }
#else
    for (int i = tid; i < LFQ_CDIM * LFQ_DIM; i += 256) sWin[i] = w_in[i];
    for (int i = tid; i < LFQ_DIM * LFQ_CDIM; i += 256) sWout[i] = w_out[i];
#endif
#if LFQ_ASYNC
    if (tid < LFQ_DIM / 4) {
        __builtin_amdgcn_global_load_async_to_lds_b128(
            (as1_i32x4*)(b_out + tid * 4),
            (as3_i32x4*)(sBout + tid * 4), 0, 0);
    }
    __builtin_amdgcn_s_wait_asynccnt(0);
#else
    for (int i = tid; i < LFQ_DIM; i += 256) sBout[i] = b_out[i];
#endif
    if (tid < LFQ_CDIM) sBin[tid] = b_in[tid];
    if (tid == 0) { sEnt = 0.0f; sCom = 0.0f; }
    __syncthreads();

    const int tl    = tid & 127;   // token within block
    const int half  = tid >> 7;    // 0 or 1
    const int token = blockIdx.x * 128 + tl;

    // ---- project_in: each half accumulates 256 of K=512 -------------------
    float acc[LFQ_CDIM];
#pragma unroll
    for (int c = 0; c < LFQ_CDIM; ++c) acc[c] = 0.0f;
    {
        const float4* xr =
            reinterpret_cast<const float4*>(x + (size_t)token * LFQ_DIM + half * 256);
        const float* wb = sWin + half * 256;
        for (int i = 0; i < 64; ++i) {
            float4 xc = xr[i];
#pragma unroll
            for (int c = 0; c < LFQ_CDIM; ++c) {
                const float4 wc =
                    *reinterpret_cast<const float4*>(wb + c * LFQ_DIM + i * 4);
                acc[c] = fmaf(xc.x, wc.x, acc[c]);
                acc[c] = fmaf(xc.y, wc.y, acc[c]);
                acc[c] = fmaf(xc.z, wc.z, acc[c]);
                acc[c] = fmaf(xc.w, wc.w, acc[c]);
            }
        }
    }
    if (half == 1) {
#pragma unroll
        for (int c = 0; c < LFQ_CDIM; ++c) sY[tl * 16 + c] = acc[c];
    }
    __syncthreads();
    if (half == 0) {
#pragma unroll
        for (int c = 0; c < LFQ_CDIM; ++c) acc[c] += sY[tl * 16 + c] + sBin[c];
#pragma unroll
        for (int c = 0; c < LFQ_CDIM; ++c) sY[tl * 16 + c] = acc[c];
    }
    __syncthreads();

    float y[LFQ_CDIM];
#pragma unroll
    for (int c = 0; c < LFQ_CDIM; ++c) y[c] = sY[tl * 16 + c];

    // ---- per-half: 7 Bernoulli dims -> entropy/commit + 128-entry prob row
    float pd[7], qd[7];
    float ent = 0.0f, commit = 0.0f;
    const int dbase = half * 7;
#pragma unroll
    for (int d = 0; d < 7; ++d) {
        const float yv = y[dbase + d];
        const bool pos = yv > 0.0f;
        const float s = pos ? 1.0f : -1.0f;
        const float df = yv - s;
        commit = fmaf(df, df, commit);
        const float t = 400.0f * yv;
        const float e = expf(-fabsf(t));       // (0,1]
        const float pmaj = 1.0f / (1.0f + e);  // prob of dominant sign
        const float pmin = e * pmaj;           // exact complement
        const float lmaj = logf(fmaxf(pmaj, 1e-20f));
        const float lmin = logf(fmaxf(pmin, 1e-20f));
        ent -= fmaf(pmaj, lmaj, pmin * lmin);  // binary entropy
        pd[d] = pos ? pmaj : pmin;             // P(bit = +1)
        qd[d] = pos ? pmin : pmaj;             // P(bit = -1)
    }
    float t3[8], t4[16];
#pragma unroll
    for (int u = 0; u < 8; ++u)
        t3[u] = (((u >> 2) & 1) ? pd[0] : qd[0]) * (((u >> 1) & 1) ? pd[1] : qd[1]) *
                ((u & 1) ? pd[2] : qd[2]);
#pragma unroll
    for (int v = 0; v < 16; ++v)
        t4[v] = (((v >> 3) & 1) ? pd[3] : qd[3]) * (((v >> 2) & 1) ? pd[4] : qd[4]) *
                (((v >> 1) & 1) ? pd[5] : qd[5]) * ((v & 1) ? pd[6] : qd[6]);

    float* prow = (half == 0 ? PL : PR) + (size_t)token * 128;
#pragma unroll
    for (int hi = 0; hi < 8; ++hi) {
        const float a = t3[hi];
#pragma unroll
        for (int lo = 0; lo < 16; lo += 4) {
            float4 vv = make_float4(a * t4[lo], a * t4[lo + 1],
                                    a * t4[lo + 2], a * t4[lo + 3]);
            *reinterpret_cast<float4*>(prow + hi * 16 + lo) = vv;
        }
    }

    // ---- index (half 0) ----------------------------------------------------
    if (half == 0) {
        int index = 0;
#pragma unroll
        for (int d = 0; d < LFQ_CDIM; ++d)
            index |= (y[d] > 0.0f ? 1 : 0) << (13 - d);
        idx_out[token] = (float)index;
    }

    // ---- project_out: each half writes 256 of 512 --------------------------
    float qv[LFQ_CDIM];
#pragma unroll
    for (int c = 0; c < LFQ_CDIM; ++c) qv[c] = y[c] > 0.0f ? 1.0f : -1.0f;
    float* orow = out + (size_t)token * LFQ_DIM + half * 256;
    const float* wob = sWout + half * 256 * LFQ_CDIM;
    const float* bob = sBout + half * 256;
    for (int d0 = 0; d0 < 256; d0 += 4) {
        float4 o;
        float* op = &o.x;
#pragma unroll
        for (int j = 0; j < 4; ++j) {
            float s = bob[d0 + j];
#pragma unroll
            for (int c = 0; c < LFQ_CDIM; ++c)
                s = fmaf(qv[c], wob[(d0 + j) * LFQ_CDIM + c], s);
            op[j] = s;
        }
        *reinterpret_cast<float4*>(orow + d0) = o;
    }

    // ---- reduce entropy / commit partials ----------------------------------
    atomicAdd(&sEnt, ent);
    atomicAdd(&sCom, commit);
    __syncthreads();
    if (tid == 0) {
        atomicAdd(&scalars[0], sEnt);
        atomicAdd(&scalars[1], sCom);
    }
}

// ------------------- kernel 2: avg_prob = PL^T * PR / N via WMMA -----------
// avg (128x128 == 16384 codes) = (1/8192) * sum_n pL_n (outer) pR_n.
// Split-K over 64 workgroups (128 tokens each); 8 waves per WG, wave w owns
// output rows [16w,16w+16) x all 128 cols = 8 accumulator tiles.
__global__ __launch_bounds__(256) void lfq_avgprob_kernel(
    const float* __restrict__ PL, const float* __restrict__ PR,
    float* __restrict__ avg) {
    const int lane = threadIdx.x & 31;
    const int wave = threadIdx.x >> 5;        // 0..7
    const int a0   = wave * 16;               // output row block
    const int m    = lane & 15;
    const int kHalf = (lane >> 4) * 2;        // K pair base: 0 or 2

    const int kChunk = LFQ_NTOK / 64;         // 128 tokens per WG
    const int n0base = blockIdx.x * kChunk;

    v8f dacc[8];
#pragma unroll
    for (int t = 0; t < 8; ++t) dacc[t] = (v8f)(0.0f);

    for (int n0 = n0base; n0 < n0base + kChunk; n0 += 4) {
        const size_t r0 = (size_t)(n0 + kHalf) * 128;
        // A tile: 16x4 slice of PL^T (rows = codeL index, K = token)
        v2f a;
        a.x = PL[r0 + a0 + m];
        a.y = PL[r0 + 128 + a0 + m];
#pragma unroll
        for (int t = 0; t < 8; ++t) {
            // B tile: 4x16 slice of PR (K = token, cols = codeR index)
            v2f b;
            b.x = PR[r0 + t * 16 + m];
            b.y = PR[r0 + 128 + t * 16 + m];
            dacc[t] = __builtin_amdgcn_wmma_f32_16x16x4_f32(
                false, a, false, b, (short)0, dacc[t], false, false);
        }
    }

    const float scale = 1.0f / (float)LFQ_NTOK;
#pragma unroll
    for (int t = 0; t < 8; ++t) {
#pragma unroll
        for (int v = 0; v < 8; ++v) {
            const int arow = a0 + v + 8 * (lane >> 4);  // C/D layout: M = v (+8 hi half)
            const int bcol = t * 16 + (lane & 15);
            atomicAdd(&avg[arow * 128 + bcol], dacc[t][v] * scale);
        }
    }
}

// ------------------- kernel 3: codebook entropy + loss combine -------------
__global__ __launch_bounds__(256) void lfq_finalize_kernel(
    const float* __restrict__ avg, const float* __restrict__ scalars,
    float* __restrict__ aux_out) {
    __shared__ float red[256];
    const int tid = threadIdx.x;
    float h = 0.0f;
    for (int i = tid; i < LFQ_K; i += 256) {
        const float p = avg[i];
        h -= p * logf(fmaxf(p, 1e-20f));
    }
    red[tid] = h;
    __syncthreads();
    for (int s = 128; s > 0; s >>= 1) {
        if (tid < s) red[tid] += red[tid + s];
        __syncthreads();
    }
    if (tid == 0) {
        const float per_sample_entropy = scalars[0] / (float)LFQ_NTOK;
        const float commit_loss = scalars[1] / (float)(LFQ_NTOK * LFQ_CDIM);
        const float entropy_aux = per_sample_entropy - 1.0f * red[0];  // gamma=1
        aux_out[0] = entropy_aux * 0.1f + commit_loss * 0.25f;
    }
}

// ---------------------------------------------------------------------------
extern "C" void kernel_launch(void* const* d_in, const int* in_sizes, int n_in,
                              void* d_out, int out_size, void* d_ws, size_t ws_size,
                              hipStream_t stream) {
    (void)in_sizes; (void)n_in; (void)out_size; (void)ws_size;
    const float* x     = (const float*)d_in[0];
    const float* w_in  = (const float*)d_in[1];
    const float* b_in  = (const float*)d_in[2];
    const float* w_out = (const float*)d_in[3];
    const float* b_out = (const float*)d_in[4];

    float* out     = (float*)d_out;
    float* idx_out = out + (size_t)LFQ_NTOK * LFQ_DIM;   // 4,194,304
    float* aux_out = idx_out + LFQ_NTOK;                 // +8192

    float* ws      = (float*)d_ws;
    float* avg     = ws;                                  // 16384
    float* scalars = ws + LFQ_K;                          // 2 (+pad to 16)
    float* PL      = ws + LFQ_K + 16;                     // 8192*128
    float* PR      = PL + (size_t)LFQ_NTOK * 128;         // 8192*128

    lfq_zero_kernel<<<(LFQ_K + 2 + 255) / 256, 256, 0, stream>>>(ws);
    lfq_main_kernel<<<LFQ_NTOK / 128, 256, 0, stream>>>(
        x, w_in, b_in, w_out, b_out, out, idx_out, PL, PR, scalars);
    lfq_avgprob_kernel<<<64, 256, 0, stream>>>(PL, PR, avg);
    lfq_finalize_kernel<<<1, 256, 0, stream>>>(avg, scalars, aux_out);
}